// Readout_11819749998810
// MI455X (gfx1250) — compile-verified
//
#include <hip/hip_runtime.h>
#include <hip/hip_bf16.h>

// ---------------------------------------------------------------------------
// MI455X (gfx1250) fused readout:
//   h  = (x@W1) * sigmoid(x@W2 + b2)          f32 GEMM -> V_WMMA_F32_16X16X4_F32
//   Z1 = segsum(h), Z2 = segmean(h), Z3 = segmax(h)   (node2graph sorted)
//   out = [Z1|Z2|Z3|gx] @ W3 + b3
// h never goes to HBM. 128-row blocks; W1/W2 K-panels double-buffered in LDS
// via GLOBAL_LOAD_ASYNC_TO_LDS_B128 (ASYNCcnt) so staging overlaps compute.
// ---------------------------------------------------------------------------

typedef __attribute__((ext_vector_type(2))) float v2f;
typedef __attribute__((ext_vector_type(8))) float v8f;

#define DV    256          // x feature dim (K of GEMM1)
#define HV    256          // hidden dim
#define GV    128          // global feature dim
#define OV    256          // output dim
#define NGR   8192         // graphs
#define KZV   (3*HV + GV)  // 896, K of GEMM2
#define ROWS2 128          // rows per block, kernel 2
#define RT2   (ROWS2/16)   // 8 row tiles per wave
#define ROWS3 32           // rows per block, kernel 3
#define LXS   (DV + 4)     // padded x-tile row stride (260)
#define ZTS   (KZV + 4)    // padded z-tile row stride (900)
#define PANEL (32 * HV)    // one 32x256 K-panel, in floats

// D = A(16x4) * B(4x16) + C, all f32.
__device__ __forceinline__ v8f wmma4(v2f a, v2f b, v8f c) {
  return __builtin_amdgcn_wmma_f32_16x16x4_f32(false, a, false, b,
                                               (short)0, c, false, false);
}

// monotone float <-> uint order maps (exact float max via atomicMax on u32)
__device__ __forceinline__ unsigned ford(float f) {
  unsigned u = __float_as_uint(f);
  return (u & 0x80000000u) ? ~u : (u | 0x80000000u);
}
__device__ __forceinline__ float ord2f(unsigned k) {
  unsigned u = (k & 0x80000000u) ? (k ^ 0x80000000u) : ~k;
  return __uint_as_float(u);
}

// flat shared pointer -> 32-bit LDS byte offset (low 32 bits of the aperture)
__device__ __forceinline__ unsigned lds_off(const void* p) {
  return (unsigned)(uintptr_t)p;
}

// Async VMEM->LDS copy of one W1 panel + one W2 panel (32x256 f32 each).
// GVS mode: lds-addr VGPR, 32-bit byte-offset VGPR, 64-bit SGPR base.
// 2048 chunks of 16B per matrix over 512 threads -> 4 b128 issues each.
__device__ __forceinline__ void stage_async(const float* gW1, const float* gW2,
                                            float* dst, int tid) {
  unsigned l1 = lds_off(dst);
  unsigned l2 = l1 + PANEL * 4;
#pragma unroll
  for (int j = 0; j < 4; ++j) {
    unsigned off = (unsigned)(tid + j * 512) * 16u;
    asm volatile("global_load_async_to_lds_b128 %0, %1, %2"
                 :: "v"(l1 + off), "v"(off), "s"(gW1) : "memory");
    asm volatile("global_load_async_to_lds_b128 %0, %1, %2"
                 :: "v"(l2 + off), "v"(off), "s"(gW2) : "memory");
  }
}
__device__ __forceinline__ void wait_async0() {
  asm volatile("s_wait_asynccnt 0" ::: "memory");
}

// ---------------------------------------------------------------------------
// Kernel 1: zero the workspace (harness poisons d_ws each run)
// ---------------------------------------------------------------------------
__global__ void rd_init(float* __restrict__ Z1, unsigned* __restrict__ Z3o,
                        float* __restrict__ cnt, int n) {
  int i = blockIdx.x * blockDim.x + threadIdx.x;
  if (i < n) { Z1[i] = 0.0f; Z3o[i] = 0u; }
  if (i < NGR) cnt[i] = 0.0f;
}

// ---------------------------------------------------------------------------
// Kernel 2: gated GEMM + fused segment sum/max/count.
// grid = V/128 blocks, block = 512 threads (16 waves).
// Wave w owns columns [16w,16w+16), all 8 row tiles, both matrices:
// 16 WMMA accumulator chains (128 acc VGPRs/lane).
// ---------------------------------------------------------------------------
__global__ void __launch_bounds__(512)
rd_gated_gemm(const float* __restrict__ x, const int* __restrict__ n2g,
              const float* __restrict__ W1, const float* __restrict__ W2,
              const float* __restrict__ b2,
              float* __restrict__ Z1, unsigned* __restrict__ Z3o,
              float* __restrict__ cnt) {
  extern __shared__ float smem[];
  float* lx  = smem;                     // [128][260] x tile (130 KB)
  float* wb0 = smem + ROWS2 * LXS;       // panel buffer 0: W1 | W2 (64 KB)
  float* wb1 = wb0 + 2 * PANEL;          // panel buffer 1: W1 | W2 (64 KB)
  float* hsm = lx;                       // reuse x tile for h tile [128][256]
  int*   sg  = (int*)(wb1 + 2 * PANEL);  // [128] graph ids

  const int tid  = threadIdx.x;
  const int lane = tid & 31;
  const int wv   = tid >> 5;             // 0..15
  const int hlf  = lane >> 4;            // half-wave select
  const int l15  = lane & 15;
  const int row0 = blockIdx.x * ROWS2;
  const int colb = wv * 16;

  // kick off async staging of the first W panel pair immediately
  stage_async(W1, W2, wb0, tid);

  // stage x tile (128x256 f32 = 128 KB) and graph ids
  {
    const float4* gx4 = (const float4*)(x + (size_t)row0 * DV);
    for (int i = tid; i < ROWS2 * (DV / 4); i += 512) {
      int r = i >> 6, c4 = i & 63;
      *(float4*)(lx + r * LXS + c4 * 4) = gx4[r * 64 + c4];
    }
    if (tid < ROWS2) sg[tid] = n2g[row0 + tid];
  }
  wait_async0();
  __syncthreads();
  if (tid < ROWS2) atomicAdd(&cnt[sg[tid]], 1.0f);   // each row counted once

  v8f acc[2][RT2] = {};   // [matrix 0=W1,1=W2][row tile]

  for (int kpi = 0; kpi < DV / 32; ++kpi) {
    float* cw  = (kpi & 1) ? wb1 : wb0;   // compute buffer
    float* nw  = (kpi & 1) ? wb0 : wb1;   // staging buffer
    if (kpi < DV / 32 - 1) {
      const int kp = (kpi + 1) * 32;
      stage_async(W1 + (size_t)kp * HV, W2 + (size_t)kp * HV, nw, tid);
    }
    float* w1s = cw;
    float* w2s = cw + PANEL;
#pragma unroll
    for (int k0 = 0; k0 < 32; k0 += 4) {
      const int ka = kpi * 32 + k0 + 2 * hlf; // A: VGPR pair {K,K+1}; halves split K
      const int kb = k0 + hlf;                // B: halves hold K rows 0/1 then 2/3
      v2f bw1 = { w1s[kb * HV + colb + l15], w1s[(kb + 2) * HV + colb + l15] };
      v2f bw2 = { w2s[kb * HV + colb + l15], w2s[(kb + 2) * HV + colb + l15] };
#pragma unroll
      for (int t = 0; t < RT2; ++t) {
        v2f a = *(const v2f*)(lx + (t * 16 + l15) * LXS + ka);
        acc[0][t] = wmma4(a, bw1, acc[0][t]);
        acc[1][t] = wmma4(a, bw2, acc[1][t]);
      }
    }
    if (kpi < DV / 32 - 1) {
      wait_async0();      // panel issued this iteration has landed in LDS
      __syncthreads();    // and everyone is done reading the compute buffer
    }
  }
  __syncthreads();        // all waves done with lx before hsm overwrites it

  // gate: h = p * sigmoid(q + b2); write tile to LDS (reuses x tile region)
  const float bb2 = b2[colb + l15];
#pragma unroll
  for (int t = 0; t < RT2; ++t) {
#pragma unroll
    for (int r = 0; r < 8; ++r) {
      float p = acc[0][t][r];
      float q = acc[1][t][r] + bb2;
      float hv = p * (1.0f / (1.0f + __expf(-q)));
      int rowi = t * 16 + r + 8 * hlf;    // C/D layout: VGPR r -> M=r / M=r+8
      hsm[rowi * HV + colb + l15] = hv;
    }
  }
  __syncthreads();

  // run-compacted segment reduce: node2graph is sorted, so the 128 rows form
  // a few runs (~5). One thread per column walks the rows and emits one
  // atomicAdd + atomicMax per run -> ~2M atomics total on L2-resident Z.
  if (tid < HV) {
    const int c = tid;
    int g = sg[0];
    float s = 0.0f, m = -3.402823466e38f;
#pragma unroll 4
    for (int r = 0; r < ROWS2; ++r) {
      int gr = sg[r];
      if (gr != g) {
        atomicAdd(&Z1[(size_t)g * HV + c], s);
        atomicMax(&Z3o[(size_t)g * HV + c], ford(m));
        g = gr; s = 0.0f; m = -3.402823466e38f;
      }
      float hv = hsm[r * HV + c];
      s += hv;
      m = fmaxf(m, hv);
    }
    atomicAdd(&Z1[(size_t)g * HV + c], s);
    atomicMax(&Z3o[(size_t)g * HV + c], ford(m));
  }
}

// ---------------------------------------------------------------------------
// Kernel 3: finalize Z (mean, empty-segment max fix) fused into output GEMM.
// grid = NGR/32 blocks, block = 512 threads (16 waves).
// ---------------------------------------------------------------------------
__global__ void __launch_bounds__(512)
rd_out_gemm(const float* __restrict__ Z1, const unsigned* __restrict__ Z3o,
            const float* __restrict__ cnt, const float* __restrict__ gx,
            const float* __restrict__ W3, const float* __restrict__ b3,
            float* __restrict__ out) {
  extern __shared__ float smem[];
  float* zt = smem;                  // [32][900] Z tile
  float* wp = smem + ROWS3 * ZTS;    // [32][256] W3 K-panel

  const int tid  = threadIdx.x;
  const int lane = tid & 31;
  const int wv   = tid >> 5;
  const int hlf  = lane >> 4;
  const int l15  = lane & 15;
  const int g0   = blockIdx.x * ROWS3;
  const int colb = wv * 16;

  // build Z tile = [Z1 | Z1/max(cnt,1) | fixed segmax | global_x]
  for (int r = wv; r < ROWS3; r += 16) {
    const int g = g0 + r;
    const float cv  = cnt[g];
    const float inv = 1.0f / fmaxf(cv, 1.0f);
    const bool  nz  = cv > 0.0f;
    for (int c = lane; c < KZV; c += 32) {   // 256-aligned regions: no wave divergence
      float v;
      if (c < 256)      v = Z1[(size_t)g * HV + c];
      else if (c < 512) v = Z1[(size_t)g * HV + (c - 256)] * inv;
      else if (c < 768) { float mv = ord2f(Z3o[(size_t)g * HV + (c - 512)]);
                          v = nz ? mv : 0.0f; }
      else              v = gx[(size_t)g * GV + (c - 768)];
      zt[r * ZTS + c] = v;
    }
  }

  v8f acc0 = {}, acc1 = {};
  for (int kp = 0; kp < KZV; kp += 32) {
    const float4* wg = (const float4*)(W3 + (size_t)kp * OV);
    for (int i = tid; i < 2048; i += 512) ((float4*)wp)[i] = wg[i];
    __syncthreads();
#pragma unroll
    for (int k0 = 0; k0 < 32; k0 += 4) {
      const int ka = kp + k0 + 2 * hlf;
      v2f a0 = *(const v2f*)(zt + l15 * ZTS + ka);
      v2f a1 = *(const v2f*)(zt + (16 + l15) * ZTS + ka);
      const int kb = k0 + hlf;
      v2f b = { wp[kb * OV + colb + l15], wp[(kb + 2) * OV + colb + l15] };
      acc0 = wmma4(a0, b, acc0);
      acc1 = wmma4(a1, b, acc1);
    }
    __syncthreads();
  }

  const float bv = b3[colb + l15];
#pragma unroll
  for (int r = 0; r < 8; ++r) {
    const int rr = r + 8 * hlf;
    out[(size_t)(g0 + rr) * OV + colb + l15]      = acc0[r] + bv;
    out[(size_t)(g0 + 16 + rr) * OV + colb + l15] = acc1[r] + bv;
  }
}

// ---------------------------------------------------------------------------
extern "C" void kernel_launch(void* const* d_in, const int* in_sizes, int n_in,
                              void* d_out, int out_size, void* d_ws, size_t ws_size,
                              hipStream_t stream) {
  const float* x   = (const float*)d_in[0];
  const int*   n2g = (const int*)d_in[1];      // node2graph (sorted)
  const float* gx  = (const float*)d_in[2];
  const float* W1  = (const float*)d_in[3];
  const float* W2  = (const float*)d_in[4];
  const float* b2  = (const float*)d_in[5];
  const float* W3  = (const float*)d_in[6];
  const float* b3  = (const float*)d_in[7];
  float* out = (float*)d_out;

  // workspace: Z1 sums (8MB) | ordered-uint max keys (8MB) | counts (32KB)
  float*    Z1  = (float*)d_ws;
  unsigned* Z3o = (unsigned*)(Z1 + (size_t)NGR * HV);
  float*    cnt = (float*)(Z3o + (size_t)NGR * HV);

  const int V = in_sizes[0] / DV;     // 262144 (multiple of 128)
  const int nz = NGR * HV;

  rd_init<<<(nz + 255) / 256, 256, 0, stream>>>(Z1, Z3o, cnt, nz);

  const size_t lds2 = (ROWS2 * LXS + 4 * PANEL) * sizeof(float)
                    + ROWS2 * sizeof(int);                      // ~259 KB
  rd_gated_gemm<<<V / ROWS2, 512, lds2, stream>>>(x, n2g, W1, W2, b2, Z1, Z3o, cnt);

  const size_t lds3 = (ROWS3 * ZTS + 32 * OV) * sizeof(float);  // ~148 KB
  rd_out_gemm<<<NGR / ROWS3, 512, lds3, stream>>>(Z1, Z3o, cnt, gx, W3, b3, out);
}